// Anchors_39238821216330
// MI455X (gfx1250) — compile-verified
//
#include <hip/hip_runtime.h>

#ifndef __has_builtin
#define __has_builtin(x) 0
#endif

// ---------------------------------------------------------------------------
// RetinaNet-style anchor target assignment for MI455X (gfx1250, wave32).
//
// Key optimizations (reasoned from MI455X specs, compile-only environment):
//  * Feature maps are never read (reference only uses their shapes) ->
//    saves ~22 MB of useless HBM traffic; kernel is bound by the 15.7 MB of
//    output stores (~0.7us floor at 23.3 TB/s).
//  * Anchors are computed analytically per-thread from the flat anchor index
//    (level decode + grid decode + constant scale/ratio tables) -> zero
//    memory traffic for the 48,960-anchor geometry.
//  * Per-batch GT tile (64 boxes x 4 f32 = 1 KB) is DMA'd into LDS with the
//    CDNA5 Tensor Data Mover (tensor_load_to_lds + s_wait_tensorcnt), then
//    broadcast to all lanes via same-address ds_load_b128 (bank broadcast).
//  * IoU loop over 64 GT boxes unrolled x8; argmax keeps first max (strict >)
//    matching jnp.argmax semantics.
//  * One global_store_b128 (loc deltas) + one b32 (label) per anchor.
// ---------------------------------------------------------------------------

typedef unsigned int u32x4 __attribute__((ext_vector_type(4)));
typedef int          i32x8 __attribute__((ext_vector_type(8)));
typedef int          i32x4 __attribute__((ext_vector_type(4)));

#define NA 48960   // total anchors: 9*(64^2+32^2+16^2+8^2)
#define NB 16      // batch
#define NM 64      // GT boxes per image

__global__ __launch_bounds__(256) void anchor_target_kernel(
    const float* __restrict__ gt_boxes,   // [B, 64, 4] xyxy
    const int*   __restrict__ gt_labels,  // [B, 64]
    float*       __restrict__ out_loc,    // [B, A, 4]
    float*       __restrict__ out_cls)    // [B, A]
{
    __shared__ __align__(16) float sbox[NM * 4]; // x0 y0 x1 y1 per box
    __shared__ float sarea[NM];
    __shared__ float slbl[NM];

    const int tid = (int)threadIdx.x;
    const int b   = (int)blockIdx.y;
    const float* gtb = gt_boxes + (size_t)b * (NM * 4);

    // ---- Stage GT boxes for this batch into LDS -------------------------
#if defined(__gfx1250__) && __has_builtin(__builtin_amdgcn_tensor_load_to_lds) && __has_builtin(__builtin_amdgcn_s_wait_tensorcnt)
    if (tid < 32) { // single wave issues the TDM DMA (wave-level op)
        unsigned long long ga = (unsigned long long)(uintptr_t)gtb;
        unsigned lds_off = (unsigned)(uintptr_t)(&sbox[0]); // flat LDS addr low 32b == LDS offset
        // D# group 0: count=1 (valid user descriptor), lds_addr, global_addr, type=2
        u32x4 g0;
        g0[0] = 1u;
        g0[1] = lds_off;
        g0[2] = (unsigned)ga;
        g0[3] = ((unsigned)(ga >> 32) & 0x01FFFFFFu) | (2u << 30);
        // D# group 1: data_size=4B (2), tensor_dim0=256 elems, tensor_dim1=1,
        //             tile_dim0=256, tile_dim1=1, tensor_dim0_stride=256
        i32x8 g1;
        g1[0] = 0x20000;            // wg_mask=0, data_size=2 (4 bytes)
        g1[1] = (int)(256u << 16);  // tensor_dim0[15:0] in [31:16]
        g1[2] = (int)(1u << 16);    // tensor_dim0 hi=0 | tensor_dim1 lo=1
        g1[3] = (int)(256u << 16);  // tensor_dim1 hi=0 | tile_dim0=256
        g1[4] = 1;                  // tile_dim1=1, tile_dim2=0
        g1[5] = 256;                // tensor_dim0_stride (48b) low dword
        g1[6] = 0;
        g1[7] = 0;
        i32x4 z4 = {0, 0, 0, 0};    // groups 2/3 unused (<=2D tensor)
#if defined(__clang_major__) && (__clang_major__ >= 23)
        i32x8 z8 = {0, 0, 0, 0, 0, 0, 0, 0};
        __builtin_amdgcn_tensor_load_to_lds(g0, g1, z4, z4, z8, 0);
#else
        __builtin_amdgcn_tensor_load_to_lds(g0, g1, z4, z4, 0);
#endif
        __builtin_amdgcn_s_wait_tensorcnt(0); // DMA complete before barrier signal
    }
#else
    if (tid < NM * 4) sbox[tid] = gtb[tid];
#endif
    __syncthreads();

    // Precompute GT areas + labels (once per block)
    if (tid < NM) {
        float x0 = sbox[tid * 4 + 0], y0 = sbox[tid * 4 + 1];
        float x1 = sbox[tid * 4 + 2], y1 = sbox[tid * 4 + 3];
        sarea[tid] = (x1 - x0) * (y1 - y0);
        slbl[tid]  = (float)gt_labels[b * NM + tid];
    }
    __syncthreads();

    const int a = (int)blockIdx.x * 256 + tid;
    if (a >= NA) return;

    // ---- Decode anchor geometry from flat index -------------------------
    // Level layout: L0 36864 (64x64,s8,bs32) | L1 9216 (32x32,s16,bs64)
    //               L2 2304 (16x16,s32,bs128) | L3 576 (8x8,s64,bs256)
    int idx, wlog; float stride, bs;
    if (a < 36864)      { idx = a;         wlog = 6; stride = 8.0f;  bs = 32.0f;  }
    else if (a < 46080) { idx = a - 36864; wlog = 5; stride = 16.0f; bs = 64.0f;  }
    else if (a < 48384) { idx = a - 46080; wlog = 4; stride = 32.0f; bs = 128.0f; }
    else                { idx = a - 48384; wlog = 3; stride = 64.0f; bs = 256.0f; }

    const unsigned cell = (unsigned)idx / 9u;       // magic-mul, no divide
    const int k = idx - (int)cell * 9;              // anchor kind 0..8
    const int gx = (int)(cell & ((1u << wlog) - 1u));
    const int gy = (int)(cell >> wlog);

    const float cx = ((float)gx + 0.5f) * stride;
    const float cy = ((float)gy + 0.5f) * stride;

    const int ri = k / 3;        // ratio index  {0.5, 1, 2}
    const int si = k - ri * 3;   // scale index  {1, 2^(1/3), 2^(2/3)}
    const float scl  = (si == 0) ? 1.0f : ((si == 1) ? 1.2599210498948732f : 1.5874010519681994f);
    const float sqr  = (ri == 0) ? 0.70710678118654752f : ((ri == 1) ? 1.0f : 1.41421356237309505f);
    const float invr = (ri == 0) ? 2.0f : ((ri == 1) ? 1.0f : 0.5f);

    const float aw = bs * scl * sqr;   // w = bs*scale*sqrt(ratio)
    const float ah = aw * invr;        // h = w / ratio

    const float ax0 = cx - 0.5f * aw, ay0 = cy - 0.5f * ah;
    const float ax1 = cx + 0.5f * aw, ay1 = cy + 0.5f * ah;
    const float areaA = (ax1 - ax0) * (ay1 - ay0);

    // ---- IoU argmax over 64 GT boxes (LDS broadcast reads) --------------
    const float4* sb4 = (const float4*)sbox;
    float best = -1.0f;
    int   bi   = 0;
#pragma unroll 8
    for (int m = 0; m < NM; ++m) {
        float4 g = sb4[m];
        float ltx = fmaxf(ax0, g.x);
        float lty = fmaxf(ay0, g.y);
        float rbx = fminf(ax1, g.z);
        float rby = fminf(ay1, g.w);
        float iw = fmaxf(rbx - ltx, 0.0f);
        float ih = fmaxf(rby - lty, 0.0f);
        float inter = iw * ih;
        float iou = inter / (areaA + sarea[m] - inter);
        if (iou > best) { best = iou; bi = m; }   // strict > == first argmax
    }

    // ---- Labels with bg/ignore masking ----------------------------------
    float lbl = slbl[bi];
    if (best > 0.4f && best < 0.5f) lbl = -1.0f;  // ignore band
    if (best < 0.4f)                lbl = 0.0f;   // background (applied last)

    // ---- Box regression deltas vs anchor (xywh form) --------------------
    float4 gb = sb4[bi];
    float gcx = (gb.x + gb.z) * 0.5f;
    float gcy = (gb.y + gb.w) * 0.5f;
    float tx = (gcx - cx) / (aw * 0.1f);
    float ty = (gcy - cy) / (ah * 0.1f);
    float tw = __logf((gb.z - gb.x) / aw) / 0.2f;
    float th = __logf((gb.w - gb.y) / ah) / 0.2f;

    // ---- Stores: one b128 + one b32 per anchor --------------------------
    const size_t ba = (size_t)b * NA + (size_t)a;
    ((float4*)out_loc)[ba] = make_float4(tx, ty, tw, th);
    out_cls[ba] = lbl;
}

extern "C" void kernel_launch(void* const* d_in, const int* in_sizes, int n_in,
                              void* d_out, int out_size, void* d_ws, size_t ws_size,
                              hipStream_t stream) {
    (void)in_sizes; (void)n_in; (void)out_size; (void)d_ws; (void)ws_size;
    // setup_inputs order: f3,f4,f5,f6 (unused — only shapes matter), gt_boxes, gt_labels
    const float* gt_boxes  = (const float*)d_in[4];
    const int*   gt_labels = (const int*)d_in[5];
    float* out = (float*)d_out;
    float* out_loc = out;                              // B*A*4 floats
    float* out_cls = out + (size_t)NB * NA * 4;        // B*A floats

    dim3 grid((NA + 255) / 256, NB);
    anchor_target_kernel<<<grid, 256, 0, stream>>>(gt_boxes, gt_labels, out_loc, out_cls);
}